// ResConcatPool_4904852652369
// MI455X (gfx1250) — compile-verified
//
#include <hip/hip_runtime.h>
#include <hip/hip_bf16.h>
#include <stdint.h>

#define N_NODES   100000
#define N_EDGES   1600000
#define NUM_GRAPHS 512
#define NUM_FEATS 9
#define VOCAB     128
#define EMB       128
#define FINAL_DIM 384
#define OUT_DIM   128
#define SLOPE     0.2f

static_assert(N_NODES % 16 == 0, "M tiles exact");  // 100000 = 16 * 6250

typedef __attribute__((ext_vector_type(2))) float v2f;
typedef __attribute__((ext_vector_type(8))) float v8f;

// ---- order-preserving float <-> uint map (for atomic float max via atomicMax(u32)) ----
__device__ __forceinline__ unsigned f2ord(float f) {
  unsigned u = __float_as_uint(f);
  return (u & 0x80000000u) ? ~u : (u | 0x80000000u);
}
__device__ __forceinline__ float ord2f(unsigned u) {
  return (u & 0x80000000u) ? __uint_as_float(u & 0x7fffffffu) : __uint_as_float(~u);
}

// ---------------- AtomEncoder: h0[n,d] = sum_f emb[f, x[n,f], d] ----------------
__global__ void atom_encoder_k(const int* __restrict__ x,
                               const float* __restrict__ emb,
                               float* __restrict__ h0) {
  int n = blockIdx.x;
  int d = threadIdx.x;                 // 128 threads = EMB
  float acc = 0.f;
#pragma unroll
  for (int f = 0; f < NUM_FEATS; ++f) {
    int idx = x[n * NUM_FEATS + f];
    acc += emb[((size_t)f * VOCAB + idx) * EMB + d];
  }
  h0[(size_t)n * EMB + d] = acc;
}

// ---------------- fp32 WMMA GEMM: C[M x N] = A[M x K] * B[K x N] ----------------
// Block = 128 threads (4 waves). Block handles one 16-row M tile; wave w handles
// N tile (blockIdx.y*4 + w). A tile staged in LDS (row stride K+4 to dodge bank
// conflicts), shared by all 4 waves. K is compile-time -> unrolled WMMA chain.
template <int K>
__global__ __launch_bounds__(128) void gemm_wmma_k(
    const float* __restrict__ A, int lda,
    const float* __restrict__ B, int ldb,   // ldb == N (B row-major K x N)
    float* __restrict__ C, int ldc) {
  constexpr int LDSA = K + 4;               // padded row stride (floats)
  __shared__ float As[16 * LDSA];

  const int tid  = threadIdx.x;
  const int wave = tid >> 5;
  const int lane = tid & 31;
  const int half = lane >> 4;               // 0: lanes 0-15, 1: lanes 16-31
  const int l    = lane & 15;
  const int tm   = blockIdx.x;
  const int tn   = blockIdx.y * 4 + wave;

  // cooperative A-tile load: 16 rows x K cols, float4 per lane, 8 lanes per row
  {
    const float* Abase = A + (size_t)tm * 16 * lda;
    const int r = tid >> 3;                 // 0..15
    const float* Ar = Abase + (size_t)r * lda;
    float* Ls = &As[r * LDSA];
#pragma unroll
    for (int c = (tid & 7); c < (K >> 2); c += 8)
      *(float4*)(&Ls[c * 4]) = *(const float4*)(Ar + c * 4);
  }
  __syncthreads();

  const int col = tn * 16 + l;
  const float* Bp   = B + (size_t)(half * 2) * ldb + col;
  const float* Arow = &As[l * LDSA + half * 2];

  v8f acc = {};
#pragma unroll 4
  for (int k = 0; k < K; k += 4) {
    v2f a, b;
    // A 16x4: lanes0-15 -> K=k,k+1 ; lanes16-31 -> K=k+2,k+3 (consecutive pair -> b64)
    a.x = Arow[0];
    a.y = Arow[1];
    // B 4x16: VGPR0 rows k+2*half, VGPR1 rows k+1+2*half
    b.x = Bp[0];
    b.y = Bp[ldb];
    acc = __builtin_amdgcn_wmma_f32_16x16x4_f32(
        false, a, false, b, (short)0, acc, false, false);
    Arow += 4;
    Bp += (size_t)4 * ldb;
  }
  // C/D: VGPR i -> row i (lanes0-15) / row i+8 (lanes16-31)
  float* Cp = C + (size_t)(tm * 16 + half * 8) * ldc + col;
#pragma unroll
  for (int i = 0; i < 8; ++i)
    Cp[(size_t)i * ldc] = acc[i];
}

// --------------- per-node attention dots: al=T@a_s, ar=T@a_d (wave/node) ---------------
template <int DOUT>
__global__ void att_k(const float* __restrict__ T,
                      const float* __restrict__ as_, const float* __restrict__ ad_,
                      float* __restrict__ al, float* __restrict__ ar, int M) {
  int gid  = blockIdx.x * blockDim.x + threadIdx.x;
  int wid  = gid >> 5;
  int lane = gid & 31;
  if (wid >= M) return;
  float s1 = 0.f, s2 = 0.f;
#pragma unroll
  for (int d = lane; d < DOUT; d += 32) {
    float v = T[(size_t)wid * DOUT + d];
    s1 += v * as_[d];
    s2 += v * ad_[d];
  }
#pragma unroll
  for (int off = 16; off; off >>= 1) {
    s1 += __shfl_xor(s1, off, 32);
    s2 += __shfl_xor(s2, off, 32);
  }
  if (lane == 0) { al[wid] = s1; ar[wid] = s2; }
}

// --------------- per-layer scratch init: AGG=bias, mord=0, denom=0 ---------------
__global__ void init_layer_k(float* __restrict__ agg, const float* __restrict__ bias,
                             int dout, unsigned* __restrict__ mord,
                             float* __restrict__ denom, int M) {
  int i = blockIdx.x * blockDim.x + threadIdx.x;
  int tot = M * dout;
  if (i < tot) agg[i] = bias[i & (dout - 1)];   // dout is 64 or 128 (pow2)
  if (i < M) { mord[i] = 0u; denom[i] = 0.f; }
}

// --------------- edge pass A: e = leaky_relu(al[src]+ar[dst]); seg-max ---------------
__global__ void edge_logits_k(const int* __restrict__ ei, int E, int M,
                              const float* __restrict__ al, const float* __restrict__ ar,
                              float* __restrict__ ebuf, unsigned* __restrict__ mord) {
  int i = blockIdx.x * blockDim.x + threadIdx.x;
  int ET = E + M;
  if (i >= ET) return;
  int s, d;
  if (i < E) { s = ei[i]; d = ei[E + i]; } else { s = d = i - E; }
  float v = al[s] + ar[d];
  float e = v > 0.f ? v : SLOPE * v;
  ebuf[i] = e;
  atomicMax(&mord[d], f2ord(e));
}

// --------------- edge pass B: w = exp(e - m[dst]); seg-sum ---------------
__global__ void edge_exp_k(const int* __restrict__ ei, int E, int M,
                           const float* __restrict__ ebuf,
                           const unsigned* __restrict__ mord,
                           float* __restrict__ wbuf, float* __restrict__ denom) {
  int i = blockIdx.x * blockDim.x + threadIdx.x;
  int ET = E + M;
  if (i >= ET) return;
  int d = (i < E) ? ei[E + i] : (i - E);
  float w = __expf(ebuf[i] - ord2f(mord[d]));
  wbuf[i] = w;
  atomicAdd(&denom[d], w);
}

// --------------- edge pass C: AGG[dst] += (w/denom[dst]) * T[src]  (wave/edge) ---------------
template <int DOUT>
__global__ void edge_agg_k(const int* __restrict__ ei, int E, int M,
                           const float* __restrict__ wbuf, const float* __restrict__ denom,
                           const float* __restrict__ T, float* __restrict__ agg) {
  int gid  = blockIdx.x * blockDim.x + threadIdx.x;
  int wid  = gid >> 5;
  int lane = gid & 31;
  int ET = E + M;
  if (wid >= ET) return;
  int s, d;
  if (wid < E) { s = ei[wid]; d = ei[E + wid]; } else { s = d = wid - E; }
  float coef = wbuf[wid] / denom[d];
  const float* Ts = T + (size_t)s * DOUT;
  float* Ag = agg + (size_t)d * DOUT;
#pragma unroll
  for (int c = lane; c < DOUT; c += 32)
    atomicAdd(&Ag[c], Ts[c] * coef);
}

// --------------- write relu(AGG) into column slice [off, off+dout) of H[N,384] ---------------
__global__ void write_slice_k(const float* __restrict__ agg, int dout,
                              float* __restrict__ H, int off, int M) {
  int i = blockIdx.x * blockDim.x + threadIdx.x;
  if (i >= M * dout) return;
  int n = i / dout, c = i - n * dout;
  H[(size_t)n * FINAL_DIM + off + c] = fmaxf(agg[i], 0.f);
}

// --------------- pooling scratch zero ---------------
__global__ void pool_zero_k(float* __restrict__ padd, unsigned* __restrict__ pmax,
                            float* __restrict__ counts) {
  int i = blockIdx.x * blockDim.x + threadIdx.x;
  if (i < NUM_GRAPHS * FINAL_DIM) { padd[i] = 0.f; pmax[i] = 0u; }
  if (i < NUM_GRAPHS) counts[i] = 0.f;
}

__global__ void pool_count_k(const int* __restrict__ batch, int M, float* __restrict__ counts) {
  int i = blockIdx.x * blockDim.x + threadIdx.x;
  if (i < M) atomicAdd(&counts[batch[i]], 1.0f);
}

// --------------- pooling scatter (wave per node) ---------------
__global__ void pool_scatter_k(const float* __restrict__ H, const int* __restrict__ batch,
                               float* __restrict__ padd, unsigned* __restrict__ pmax, int M) {
  int gid  = blockIdx.x * blockDim.x + threadIdx.x;
  int wid  = gid >> 5;
  int lane = gid & 31;
  if (wid >= M) return;
  int g = batch[wid];
#pragma unroll
  for (int d = lane; d < FINAL_DIM; d += 32) {
    float v = H[(size_t)wid * FINAL_DIM + d];
    atomicAdd(&padd[g * FINAL_DIM + d], v);
    atomicMax(&pmax[g * FINAL_DIM + d], f2ord(v));
  }
}

// --------------- final: out[g,:] = [mean|add|max] @ lin_w + lin_b ---------------
__global__ __launch_bounds__(OUT_DIM) void final_linear_k(
    const float* __restrict__ padd, const unsigned* __restrict__ pmax,
    const float* __restrict__ counts,
    const float* __restrict__ lw, const float* __restrict__ lb,
    float* __restrict__ out) {
  __shared__ float gv[3 * FINAL_DIM];
  int g = blockIdx.x, t = threadIdx.x;   // 128 threads
  float cnt = counts[g];
  float inv = 1.0f / fmaxf(cnt, 1.0f);
  for (int d = t; d < 3 * FINAL_DIM; d += OUT_DIM) {
    float v;
    if (d < FINAL_DIM)            v = padd[g * FINAL_DIM + d] * inv;
    else if (d < 2 * FINAL_DIM)   v = padd[g * FINAL_DIM + d - FINAL_DIM];
    else                          v = (cnt > 0.f) ? ord2f(pmax[g * FINAL_DIM + d - 2 * FINAL_DIM]) : 0.f;
    gv[d] = v;
  }
  __syncthreads();
  float acc = lb[t];
  for (int d = 0; d < 3 * FINAL_DIM; ++d)
    acc += gv[d] * lw[(size_t)d * OUT_DIM + t];
  out[(size_t)g * OUT_DIM + t] = acc;
}

// =====================================================================================
extern "C" void kernel_launch(void* const* d_in, const int* in_sizes, int n_in,
                              void* d_out, int out_size, void* d_ws, size_t ws_size,
                              hipStream_t stream) {
  // setup_inputs() dict order, lists flattened element-wise:
  // 0:x 1:edge_index 2:batch 3:atom_emb 4-7:Ws 8-11:atts_src 12-15:atts_dst
  // 16-19:biases 20:lin_w 21:lin_b
  const int*   x        = (const int*)  d_in[0];
  const int*   ei       = (const int*)  d_in[1];
  const int*   batch    = (const int*)  d_in[2];
  const float* atom_emb = (const float*)d_in[3];
  const float* Ws[4]    = {(const float*)d_in[4],  (const float*)d_in[5],
                           (const float*)d_in[6],  (const float*)d_in[7]};
  const float* As[4]    = {(const float*)d_in[8],  (const float*)d_in[9],
                           (const float*)d_in[10], (const float*)d_in[11]};
  const float* Ad[4]    = {(const float*)d_in[12], (const float*)d_in[13],
                           (const float*)d_in[14], (const float*)d_in[15]};
  const float* Bs[4]    = {(const float*)d_in[16], (const float*)d_in[17],
                           (const float*)d_in[18], (const float*)d_in[19]};
  const float* lin_w    = (const float*)d_in[20];
  const float* lin_b    = (const float*)d_in[21];
  float* out = (float*)d_out;

  const int M  = N_NODES;
  const int E  = N_EDGES;
  const int ET = E + M;

  // ---- bump-allocate workspace ----
  char* p = (char*)d_ws;
  auto alloc = [&](size_t bytes) -> void* {
    void* r = (void*)p;
    p += (bytes + 255) & ~(size_t)255;
    return r;
  };
  float*    H     = (float*)   alloc((size_t)M * FINAL_DIM * 4); // residual-concat ladder
  float*    H0    = (float*)   alloc((size_t)M * EMB * 4);       // atom encoder output
  float*    T     = (float*)   alloc((size_t)M * 128 * 4);       // h@W (per layer)
  float*    AGG   = (float*)   alloc((size_t)M * 128 * 4);       // aggregated output
  float*    al    = (float*)   alloc((size_t)M * 4);
  float*    ar    = (float*)   alloc((size_t)M * 4);
  unsigned* mord  = (unsigned*)alloc((size_t)M * 4);
  float*    denom = (float*)   alloc((size_t)M * 4);
  float*    ebuf  = (float*)   alloc((size_t)ET * 4);
  float*    wbuf  = (float*)   alloc((size_t)ET * 4);
  float*    padd  = (float*)   alloc((size_t)NUM_GRAPHS * FINAL_DIM * 4);
  unsigned* pmax  = (unsigned*)alloc((size_t)NUM_GRAPHS * FINAL_DIM * 4);
  float*    cnts  = (float*)   alloc((size_t)NUM_GRAPHS * 4);
  (void)ws_size; (void)in_sizes; (void)n_in; (void)out_size;

  // ---- AtomEncoder ----
  atom_encoder_k<<<M, EMB, 0, stream>>>(x, atom_emb, H0);

  // layer configs: input ptr / lda / K / dout / output column offset in H
  struct LCfg { const float* A; int lda; int K; int dout; int off; };
  LCfg L[4] = {
    {H0,       EMB,       128,  64, 320},
    {H + 320,  FINAL_DIM,  64, 128, 192},
    {H + 192,  FINAL_DIM, 192, 128,  64},
    {H +  64,  FINAL_DIM, 320,  64,   0},
  };

  for (int i = 0; i < 4; ++i) {
    int K = L[i].K, dout = L[i].dout, off = L[i].off;

    // T = A @ W   (fp32 WMMA; M = 6250 exact 16-row tiles; 4 N-tiles per block)
    dim3 ggrid(M / 16, dout / 64);
    switch (i) {
      case 0: gemm_wmma_k<128><<<ggrid, 128, 0, stream>>>(L[i].A, L[i].lda, Ws[i], dout, T, dout); break;
      case 1: gemm_wmma_k< 64><<<ggrid, 128, 0, stream>>>(L[i].A, L[i].lda, Ws[i], dout, T, dout); break;
      case 2: gemm_wmma_k<192><<<ggrid, 128, 0, stream>>>(L[i].A, L[i].lda, Ws[i], dout, T, dout); break;
      case 3: gemm_wmma_k<320><<<ggrid, 128, 0, stream>>>(L[i].A, L[i].lda, Ws[i], dout, T, dout); break;
    }

    // al / ar
    if (dout == 64)
      att_k<64><<<(M * 32 + 127) / 128, 128, 0, stream>>>(T, As[i], Ad[i], al, ar, M);
    else
      att_k<128><<<(M * 32 + 127) / 128, 128, 0, stream>>>(T, As[i], Ad[i], al, ar, M);

    // AGG = bias, mord = 0, denom = 0
    init_layer_k<<<(M * dout + 255) / 256, 256, 0, stream>>>(AGG, Bs[i], dout, mord, denom, M);

    // segment softmax over dst + aggregation
    edge_logits_k<<<(ET + 255) / 256, 256, 0, stream>>>(ei, E, M, al, ar, ebuf, mord);
    edge_exp_k   <<<(ET + 255) / 256, 256, 0, stream>>>(ei, E, M, ebuf, mord, wbuf, denom);
    if (dout == 64)
      edge_agg_k<64><<<((size_t)ET * 32 + 255) / 256, 256, 0, stream>>>(ei, E, M, wbuf, denom, T, AGG);
    else
      edge_agg_k<128><<<((size_t)ET * 32 + 255) / 256, 256, 0, stream>>>(ei, E, M, wbuf, denom, T, AGG);

    // H[:, off:off+dout] = relu(AGG)   (residual concat ladder, filled right-to-left)
    write_slice_k<<<(M * dout + 255) / 256, 256, 0, stream>>>(AGG, dout, H, off, M);
  }

  // ---- graph pooling ----
  pool_zero_k   <<<(NUM_GRAPHS * FINAL_DIM + 255) / 256, 256, 0, stream>>>(padd, pmax, cnts);
  pool_count_k  <<<(M + 255) / 256, 256, 0, stream>>>(batch, M, cnts);
  pool_scatter_k<<<((size_t)M * 32 + 255) / 256, 256, 0, stream>>>(H, batch, padd, pmax, M);

  // ---- final linear ----
  final_linear_k<<<NUM_GRAPHS, OUT_DIM, 0, stream>>>(padd, pmax, cnts, lin_w, lin_b, out);
}